// MinimalLRU_19301583028275
// MI455X (gfx1250) — compile-verified
//
#include <hip/hip_runtime.h>
#include <stdint.h>

// Problem constants (reference: B=4, L=8192, D=512)
#define BB_   4
#define LL_   8192
#define DD_   512
#define NCH_  8          // time chunks for cross-chunk seed fix-up
#define CHUNK_ (LL_ / NCH_)   // 1024
#define BLK_  256        // timesteps per WMMA batch (16 columns x 16 rows)
#define PITCH_ 17        // padded LDS row pitch (floats)

typedef float v2f __attribute__((ext_vector_type(2)));
typedef float v8f __attribute__((ext_vector_type(8)));
typedef unsigned int u32x4 __attribute__((ext_vector_type(4)));
typedef int i32x4 __attribute__((ext_vector_type(4)));
typedef int i32x8 __attribute__((ext_vector_type(8)));

// ---------------------------------------------------------------------------
// Kernel 1: per-(b, chunk, d) zero-seeded chunk end state (scalar recurrence).
// Lanes map to consecutive d -> 128B coalesced loads per timestep.
// ---------------------------------------------------------------------------
__global__ __launch_bounds__(256) void lru_chunk_end(
    const float* __restrict__ x, const float* __restrict__ nu,
    const float* __restrict__ theta, float2* __restrict__ E) {
  int id = blockIdx.x * blockDim.x + threadIdx.x;
  if (id >= BB_ * NCH_ * DD_) return;
  int d = id % DD_;
  int c = (id / DD_) % NCH_;
  int b = id / (DD_ * NCH_);
  float ed = expf(nu[d]);
  float rho = expf(-ed);
  float s, ct;
  sincosf(theta[d], &s, &ct);
  float la = rho * ct, lb = rho * s;  // lambda = la + i*lb
  const float* xp = x + ((size_t)(b * LL_ + c * CHUNK_)) * DD_ + d;
  float hr = 0.f, hi = 0.f;
#pragma unroll 4
  for (int t = 0; t < CHUNK_; ++t) {
    float xv = xp[(size_t)t * DD_];
    float nr = la * hr - lb * hi + xv;
    hi = lb * hr + la * hi;
    hr = nr;
  }
  E[(b * NCH_ + c) * DD_ + d] = make_float2(hr, hi);
}

// ---------------------------------------------------------------------------
// Kernel 2: per-(b,d) exclusive scan over NCH_ chunk states -> seeds.
// ---------------------------------------------------------------------------
__global__ __launch_bounds__(256) void lru_seed_scan(
    const float* __restrict__ nu, const float* __restrict__ theta,
    const float2* __restrict__ E, float2* __restrict__ S) {
  int id = blockIdx.x * blockDim.x + threadIdx.x;
  if (id >= BB_ * DD_) return;
  int d = id % DD_;
  int b = id / DD_;
  float ed = expf(nu[d]);
  float rho = expf(-ed);
  float s, ct;
  sincosf(theta[d], &s, &ct);
  float pr = rho * ct, pi = rho * s;
  for (int q = 0; q < 10; ++q) {  // lambda^1024 by repeated squaring
    float nr = pr * pr - pi * pi;
    pi = 2.f * pr * pi;
    pr = nr;
  }
  float hr = 0.f, hi = 0.f;
  for (int c = 0; c < NCH_; ++c) {
    S[(b * NCH_ + c) * DD_ + d] = make_float2(hr, hi);
    float2 e = E[(b * NCH_ + c) * DD_ + d];
    float nr = pr * hr - pi * hi + e.x;
    hi = pi * hr + pr * hi + e.y;
    hr = nr;
  }
}

// ---------------------------------------------------------------------------
// TDM: load a 16(t-rows) x 16(d-cols) f32 tile from the strided x tensor into
// LDS with 1-DWORD padding every 16 DWORDs (pitch-17 rows).
// D# per CDNA5 ISA §8: group0 {count,lds_addr,global_addr,type=2},
// group1 {data_size=4B, pad_enable, pad_interval=16dw, tensor_dim0=512,
//         tensor_dim1=8192, tile=16x16, dim0_stride=512, dim1_stride=L*D}.
// Toolchain here is the 6-arg builtin: (g0, g1, g2, g3, g4, cpol).
// ---------------------------------------------------------------------------
__device__ __forceinline__ void tdm_load_tile(uint32_t lds_byte, const float* gptr) {
  uint64_t ga = (uint64_t)(uintptr_t)gptr;
  u32x4 g0;
  g0[0] = 1u;                                               // count=1, user mode
  g0[1] = lds_byte;                                         // lds_addr
  g0[2] = (uint32_t)ga;                                     // global_addr lo
  g0[3] = (uint32_t)((ga >> 32) & 0x1FFFFFFu) | (2u << 30); // ga hi | type=2
  i32x8 g1;
  g1[0] = (int)((2u << 16) | (1u << 20) | (3u << 22));      // 4B, pad_en, intvl=16dw, amt=1dw
  g1[1] = (int)((uint32_t)DD_ << 16);                       // tensor_dim0 = 512
  g1[2] = (int)(((uint32_t)LL_ & 0xFFFFu) << 16);           // tensor_dim1 = 8192 (lo16)
  g1[3] = (int)(((uint32_t)LL_ >> 16) | (16u << 16));       // dim1 hi | tile_dim0=16
  g1[4] = (int)(16u);                                       // tile_dim1=16, tile_dim2=0
  g1[5] = (int)((uint32_t)DD_);                             // dim0_stride = 512
  g1[6] = (int)((((uint32_t)(LL_ * DD_)) & 0xFFFFu) << 16); // stride0 hi | stride1 lo16
  g1[7] = (int)(((uint32_t)(LL_ * DD_)) >> 16);             // stride1 hi
  i32x4 z4 = {0, 0, 0, 0};
  i32x8 z8 = {0, 0, 0, 0, 0, 0, 0, 0};
  __builtin_amdgcn_tensor_load_to_lds(g0, g1, z4, z4, z8, 0);
}

// ---------------------------------------------------------------------------
// Kernel 3: main WMMA scan. One workgroup = 16 waves = 16 channels, one
// (b, chunk). Each wave: 4 blocks of 256 timesteps; per block:
//   Dr/Di = T_re/T_im (16x16 Toeplitz, A operand) x X (16 rows x 16 tiles, B)
//   via 4 chained v_wmma_f32_16x16x4_f32 each, then a Kogge-Stone affine
//   carry scan across the 16 tile-columns, seed injection, y = gr*hr - gi*hi.
// TDM loads are per-wave 16x16 sub-tiles, double-buffered.
// ---------------------------------------------------------------------------
__global__ __launch_bounds__(512) void lru_wmma_scan(
    const float* __restrict__ x, const float* __restrict__ nu,
    const float* __restrict__ theta, const float* __restrict__ gre,
    const float* __restrict__ gim, const float2* __restrict__ seeds,
    float* __restrict__ y) {
  __shared__ float inbuf[2][BLK_ * PITCH_];   // 2 x 17KB, TDM destination
  __shared__ float stg[BLK_ * PITCH_];        // output staging

  const int d0 = blockIdx.x * 16;
  const int c = blockIdx.y;
  const int b = blockIdx.z;
  const int tid = threadIdx.x;
  const int w = tid >> 5;      // wave id = local channel
  const int l = tid & 31;      // lane
  const int j = l & 15;        // WMMA column (tile index) / row group
  const int h = l >> 4;        // lane half
  const int d = d0 + w;

  const float th = theta[d];
  const float ed = expf(nu[d]);
  const float rho = expf(-ed);
  float s0, c0;
  sincosf(th, &s0, &c0);
  const float la = rho * c0, lb = rho * s0;  // lambda
  const float gr = gre[d], gi = gim[d];

  // A operands: lower-triangular Toeplitz, Re and Im.
  // A 16x4 f32 layout: lanes hold M=l%16; VGPR v, half h -> K = 4*kk + v + 2*h.
  v2f Ar[4], Ai[4];
  for (int kk = 0; kk < 4; ++kk)
    for (int v = 0; v < 2; ++v) {
      int K = 4 * kk + v + 2 * h;
      int m = j - K;  // A-row M = l%16 = j
      float vr = 0.f, vi = 0.f;
      if (m >= 0) {
        float sm, cm;
        sincosf((float)m * th, &sm, &cm);
        float rm = expf(-(float)m * ed);
        vr = rm * cm;
        vi = rm * sm;
      }
      Ar[kk][v] = vr;
      Ai[kk][v] = vi;
    }

  // Per-lane seed powers lambda^(M+1) for D rows M = r + 8*h.
  float pr8[8], pi8[8];
  for (int r = 0; r < 8; ++r) {
    float tau = (float)(r + 8 * h + 1);
    float st, ctt;
    sincosf(tau * th, &st, &ctt);
    float rm = expf(-tau * ed);
    pr8[r] = rm * ctt;
    pi8[r] = rm * st;
  }
  // lambda^16 (tile-step factor) by squaring.
  float P16r = la, P16i = lb;
  for (int q = 0; q < 4; ++q) {
    float nr = P16r * P16r - P16i * P16i;
    P16i = 2.f * P16r * P16i;
    P16r = nr;
  }

  float2 sd = seeds[(b * NCH_ + c) * DD_ + d];
  float Cr = sd.x, Ci = sd.y;  // running carry into this chunk

  const int t0c = c * CHUNK_;
  const float* xb = x + (size_t)b * LL_ * DD_;
  const uint32_t inb0 = (uint32_t)(uintptr_t)(&inbuf[0][0]);
  const uint32_t inb1 = (uint32_t)(uintptr_t)(&inbuf[1][0]);
  const uint32_t wave_lds_off = (uint32_t)(w * 16 * PITCH_ * 4);

  // Prologue: TDM for block 0 (each wave loads its own 16-row slice).
  tdm_load_tile(inb0 + wave_lds_off, xb + (size_t)(t0c + 16 * w) * DD_ + d0);

  for (int k = 0; k < CHUNK_ / BLK_; ++k) {
    if (k + 1 < CHUNK_ / BLK_) {
      uint32_t base = ((k + 1) & 1) ? inb1 : inb0;
      tdm_load_tile(base + wave_lds_off,
                    xb + (size_t)(t0c + (k + 1) * BLK_ + 16 * w) * DD_ + d0);
      __builtin_amdgcn_s_wait_tensorcnt(1);  // block k's load complete
    } else {
      __builtin_amdgcn_s_wait_tensorcnt(0);
    }
    __syncthreads();

    const float* in = inbuf[k & 1];
    // B operand: X[s, col] = x(t = 16*col + s, channel w). Layout mirrors A:
    // VGPR v, half h -> K = 4*kk + v + 2*h ; N = l%16 = j.
    v2f Bx[4];
    for (int kk = 0; kk < 4; ++kk)
      for (int v = 0; v < 2; ++v) {
        int K = 4 * kk + v + 2 * h;
        Bx[kk][v] = in[(16 * j + K) * PITCH_ + w];
      }

    v8f Dr = {}, Di = {};
    for (int kk = 0; kk < 4; ++kk) {
      Dr = __builtin_amdgcn_wmma_f32_16x16x4_f32(false, Ar[kk], false, Bx[kk],
                                                 (short)0, Dr, false, false);
      Di = __builtin_amdgcn_wmma_f32_16x16x4_f32(false, Ai[kk], false, Bx[kk],
                                                 (short)0, Di, false, false);
    }

    // Local end states E[m] (row M=15) live in lanes 16+m of D VGPR 7.
    float dr7 = Dr[7], di7 = Di[7];
    float epr = __shfl(dr7, 16 + ((j + 15) & 15), 32);  // E[j-1]
    float epi = __shfl(di7, 16 + ((j + 15) & 15), 32);
    // Affine map per column: S_j = A*S_{j-1} + B ; S_{-1}=0.
    float mAr = (j == 0) ? 0.f : P16r;
    float mAi = (j == 0) ? 0.f : P16i;
    float mBr = (j == 0) ? Cr : epr;
    float mBi = (j == 0) ? Ci : epi;
    for (int off = 1; off < 16; off <<= 1) {
      int src = (h << 4) + ((j - off) & 15);
      float par = __shfl(mAr, src, 32), pai = __shfl(mAi, src, 32);
      float pbr = __shfl(mBr, src, 32), pbi = __shfl(mBi, src, 32);
      if (j >= off) {
        float nbr = mAr * pbr - mAi * pbi + mBr;
        float nbi = mAr * pbi + mAi * pbr + mBi;
        float nar = mAr * par - mAi * pai;
        float nai = mAr * pai + mAi * par;
        mAr = nar; mAi = nai; mBr = nbr; mBi = nbi;
      }
    }
    float Sr = mBr, Si = mBi;  // carry-in state for column j

    // y = gr*(Dr + Re(lambda^{M+1} S)) - gi*(Di + Im(lambda^{M+1} S))
    for (int r = 0; r < 8; ++r) {
      float hr = Dr[r] + pr8[r] * Sr - pi8[r] * Si;
      float hi = Di[r] + pr8[r] * Si + pi8[r] * Sr;
      int t = 16 * j + r + 8 * h;
      stg[t * PITCH_ + w] = gr * hr - gi * hi;
    }

    // Next-block carry: E[15] + lambda^16 * S[15].
    float e15r = __shfl(dr7, 31, 32), e15i = __shfl(di7, 31, 32);
    float S15r = __shfl(Sr, 15, 32), S15i = __shfl(Si, 15, 32);
    Cr = e15r + P16r * S15r - P16i * S15i;
    Ci = e15i + P16r * S15i + P16i * S15r;

    __syncthreads();
    // Cooperative coalesced store of the 256x16 output block.
    int tbase = t0c + k * BLK_;
    for (int rep = 0; rep < 8; ++rep) {
      int idx = rep * 512 + tid;
      int tt = idx >> 4, dd = idx & 15;
      y[((size_t)b * LL_ + tbase + tt) * DD_ + d0 + dd] = stg[tt * PITCH_ + dd];
    }
    __syncthreads();
  }
}

// ---------------------------------------------------------------------------
extern "C" void kernel_launch(void* const* d_in, const int* in_sizes, int n_in,
                              void* d_out, int out_size, void* d_ws, size_t ws_size,
                              hipStream_t stream) {
  const float* x = (const float*)d_in[0];
  const float* nu = (const float*)d_in[1];
  const float* theta = (const float*)d_in[2];
  const float* gre = (const float*)d_in[3];
  const float* gim = (const float*)d_in[4];
  float* y = (float*)d_out;

  float2* E = (float2*)d_ws;                       // B*NCH*D chunk end states
  float2* S = E + (size_t)BB_ * NCH_ * DD_;        // B*NCH*D seeds

  lru_chunk_end<<<(BB_ * NCH_ * DD_ + 255) / 256, 256, 0, stream>>>(x, nu, theta, E);
  lru_seed_scan<<<(BB_ * DD_ + 255) / 256, 256, 0, stream>>>(nu, theta, E, S);
  dim3 grid(DD_ / 16, NCH_, BB_);
  lru_wmma_scan<<<grid, 512, 0, stream>>>(x, nu, theta, gre, gim, S, y);
}